// RealtimeNgramProcessor_17703855194503
// MI455X (gfx1250) — compile-verified
//
#include <hip/hip_runtime.h>

// n-gram id encoder for gfx1250 (MI455X).
// Two-level sorted-table lookup:
//   level 0: every-8th key ("pivot") staged into LDS with CDNA5 async
//            global->LDS DMA (ASYNCcnt path); branchless dual binary search
//            over LDS (both ngram sizes interleaved for ILP on the ds_load
//            latency chain)
//   level 1: 32B-aligned <=8-key segment fetched as 2x b128 from L2-resident
//            table; single vals fetch on hit.
//
// Inputs (int32 on device per harness convention):
//   d_in[0] = x      [B*S]   tokens in [0,512)
//   d_in[1] = keys2  [nk2]   sorted unique packed 2-gram keys
//   d_in[2] = vals2  [nk2]
//   d_in[3] = keys3  [nk3]
//   d_in[4] = vals3  [nk3]
// Output: float [2, B, S]  (id or 0 for OOV)

#define TOKV      512
#define SEG_SHIFT 3
#define SEG       8
#define MAX_PIV   6250   // ceil(50000 / 8)
#define CHUNKS    2      // blocks per row (occupancy vs staging amortization)

// Low 32 bits of a flat pointer to __shared__ data are the wave-relative LDS
// byte offset (flat LDS addr = {shared_aperture_hi, lds_offset}).
__device__ __forceinline__ unsigned lds_off_of(const void* p) {
    return (unsigned)(unsigned long long)p;
}

// Stage keys[i*8] for i in [0, ceil(nk/8)) into lds_arr using
// GLOBAL_LOAD_ASYNC_TO_LDS_B32 (GVS mode: saddr 64-bit base + vaddr 32-bit
// per-lane byte offset). Caller must s_wait_asynccnt + barrier afterwards.
__device__ __forceinline__ void async_stage_pivots(const int* __restrict__ keys,
                                                   int nk, int* lds_arr) {
    const int np = (nk + SEG - 1) >> SEG_SHIFT;
    const unsigned base = lds_off_of(lds_arr);
    for (int i = threadIdx.x; i < np; i += blockDim.x) {
        unsigned loff = base + ((unsigned)i << 2);          // LDS byte addr
        unsigned goff = ((unsigned)i << (SEG_SHIFT + 2));   // keys[i*8] byte off
        asm volatile("global_load_async_to_lds_b32 %0, %1, %2"
                     :
                     : "v"(loff), "v"(goff), "s"(keys)
                     : "memory");
    }
}

// Level-1: exact-match scan of segment [seg*8, min(seg*8+8, nk)).
__device__ __forceinline__ int seg_scan(int v, int seg,
                                        const int* __restrict__ keys,
                                        const int* __restrict__ vals, int nk) {
    if (seg < 0) return 0;                    // v < keys[0]
    const int start = seg << SEG_SHIFT;
    int hit = -1;
    if (start + SEG <= nk) {
        // segment start is 8-int (32B) aligned -> two b128 loads
        const int4* k4 = reinterpret_cast<const int4*>(keys + start);
        const int4 a = k4[0];
        const int4 b = k4[1];
        if (a.x == v) hit = start + 0;
        if (a.y == v) hit = start + 1;
        if (a.z == v) hit = start + 2;
        if (a.w == v) hit = start + 3;
        if (b.x == v) hit = start + 4;
        if (b.y == v) hit = start + 5;
        if (b.z == v) hit = start + 6;
        if (b.w == v) hit = start + 7;
    } else {                                  // partial tail segment only
        for (int j = start; j < nk; ++j)
            if (keys[j] == v) hit = j;
    }
    return (hit >= 0) ? vals[hit] : 0;
}

__global__ __launch_bounds__(256)
void RealtimeNgramProcessor_17703855194503_kernel(
    const int* __restrict__ x,
    const int* __restrict__ keys2, const int* __restrict__ vals2, int nk2,
    const int* __restrict__ keys3, const int* __restrict__ vals3, int nk3,
    float* __restrict__ out, int S)
{
    __shared__ int pk2[MAX_PIV];
    __shared__ int pk3[MAX_PIV];

    async_stage_pivots(keys2, nk2, pk2);
    async_stage_pivots(keys3, nk3, pk3);
    asm volatile("s_wait_asynccnt 0" ::: "memory");  // my wave's DMAs done
    __syncthreads();                                  // whole block's DMAs done

    const int np2 = (nk2 + SEG - 1) >> SEG_SHIFT;
    const int np3 = (nk3 + SEG - 1) >> SEG_SHIFT;

    const int b = blockIdx.x;                         // row
    const int span = S / CHUNKS;
    const int s0 = blockIdx.y * span;                 // chunk of the row
    const long long BS = (long long)gridDim.x * S;
    const int* xrow = x + (long long)b * S;
    float* o2 = out + (long long)b * S;               // n=2 plane
    float* o3 = out + BS + (long long)b * S;          // n=3 plane

    for (int s = s0 + threadIdx.x; s < s0 + span; s += blockDim.x) {
        int t0 = xrow[s];
        int t1 = (s >= 1) ? xrow[s - 1] : 0;          // left zero-padding
        int t2 = (s >= 2) ? xrow[s - 2] : 0;
        int v2 = t1 * TOKV + t0;
        int v3 = (t2 * TOKV + t1) * TOKV + t0;

        // Branchless dual binary search: after the loop, loN = number of
        // pivots <= vN (classic lower-bound skeleton; np < 2^13 = 8192).
        // Both chains interleave -> 2 independent ds_loads per step.
        int lo2 = 0, lo3 = 0;
        #pragma unroll
        for (int step = 4096; step > 0; step >>= 1) {
            int t2i = lo2 + step;
            int t3i = lo3 + step;
            int i2 = ((t2i <= np2) ? t2i : np2) - 1;  // always valid index
            int i3 = ((t3i <= np3) ? t3i : np3) - 1;
            int q2 = pk2[i2];
            int q3 = pk3[i3];
            if ((t2i <= np2) & (q2 <= v2)) lo2 = t2i;
            if ((t3i <= np3) & (q3 <= v3)) lo3 = t3i;
        }

        o2[s] = (float)seg_scan(v2, lo2 - 1, keys2, vals2, nk2);
        o3[s] = (float)seg_scan(v3, lo3 - 1, keys3, vals3, nk3);
    }
}

extern "C" void kernel_launch(void* const* d_in, const int* in_sizes, int n_in,
                              void* d_out, int out_size, void* d_ws, size_t ws_size,
                              hipStream_t stream) {
    (void)n_in; (void)d_ws; (void)ws_size; (void)out_size;
    const int* x     = (const int*)d_in[0];
    const int* keys2 = (const int*)d_in[1];
    const int* vals2 = (const int*)d_in[2];
    const int* keys3 = (const int*)d_in[3];
    const int* vals3 = (const int*)d_in[4];
    float* out = (float*)d_out;

    const int S = 8192;
    const int B = in_sizes[0] / S;   // 256
    const int nk2 = in_sizes[1];
    const int nk3 = in_sizes[3];

    dim3 grid(B, CHUNKS), block(256);
    RealtimeNgramProcessor_17703855194503_kernel<<<grid, block, 0, stream>>>(
        x, keys2, vals2, nk2, keys3, vals3, nk3, out, S);
}